// DiffractiveLayer_19361712571162
// MI455X (gfx1250) — compile-verified
//
#include <hip/hip_runtime.h>
#include <math.h>

// ---------------------------------------------------------------------------
// Diffractive layer: out[j] = dA * sum_i modes[i] * g(r_ij)
// g depends only on (ix-jx, iy-jy)  ->  191x191 table, then a 2D complex
// correlation done as per-dy Toeplitz GEMMs on V_WMMA_F32_16X16X4_F32.
//
// v2 changes vs v1 (from asm inspection):
//  - zero-padded modes (96 x 286 cols) -> no exec-divergent guard in the
//    inner loop (removes s_and_saveexec / cbranch per iteration)
//  - row-pair interleaved modes layout -> B fragment is ONE coalesced,
//    8B-aligned global_load_b64 per lane (was 4x b32)
//  - #pragma unroll 4 for deeper load/WMMA pipelining
// ---------------------------------------------------------------------------

#define NN      96
#define M_TOT   (NN * NN)        // 9216
#define DMAX    191              // 2N-1
#define TABLE   (DMAX * DMAX)    // 36481
#define PADW    286              // 96 + 2*95 padded columns
#define PAIRS   48               // 96/2 row pairs
#define PROW    (PADW * 2)       // 572 floats per row-pair
#define TDIM    6                // 96/16 tiles per dimension
#define NTILE   (TDIM * TDIM)    // 36 output tiles
#define SLICES  16               // split-K over dy diagonals
#define DY_PER  12               // ceil(191/16)

typedef float v2f __attribute__((ext_vector_type(2)));
typedef float v8f __attribute__((ext_vector_type(8)));

// workspace float layout:
//   [0,       TABLE)      g_re   (indexed [dyi*191 + dxi])
//   [TABLE,   2*TABLE)    g_im
//   g2 = 2*TABLE:
//   [g2,          g2+27456)   m_re2: [(k/2)*572 + col*2 + (k&1)], col in [0,286)
//   [g2+27456,    g2+54912)   m_im2
//   then partials: ((slice*NTILE + tile)*2 + comp)*256 + elem
#define WS_GRE   0
#define WS_GIM   TABLE
#define WS_MRE   (2 * TABLE)
#define WS_MIM   (2 * TABLE + PAIRS * PROW)
#define WS_PART  (2 * TABLE + 2 * PAIRS * PROW)

// --------------------------- setup: table + modes ---------------------------
__global__ void dol_setup_kernel(const float* __restrict__ x,
                                 const float* __restrict__ w,
                                 float* __restrict__ ws)
{
    int idx = blockIdx.x * blockDim.x + threadIdx.x;

    float* g_re  = ws + WS_GRE;
    float* g_im  = ws + WS_GIM;
    float* m_re2 = ws + WS_MRE;
    float* m_im2 = ws + WS_MIM;

    if (idx < TABLE) {
        int dxi = idx % DMAX;
        int dyi = idx / DMAX;
        float dx = (float)(dxi - 95) * 1.55e-6f;
        float dy = (float)(dyi - 95) * 1.55e-6f;
        const float dz = 1e-5f;
        float r2 = dx * dx + dy * dy + dz * dz;
        float r  = sqrtf(r2);
        float amp = dz / r2;                              // dz / r^2
        float a   = 1.0f / (6.28318530717958647692f * r); // 1/(2*pi*r)
        const float invlam = 1.0f / 1.55e-6f;             // 1/lambda
        const float kwav   = 6.28318530717958647692f / 1.55e-6f;
        float s, c;
        sincosf(kwav * r, &s, &c);
        // (a - j*invlam) * (c + j*s) * amp
        g_re[idx] = amp * (a * c + invlam * s);
        g_im[idx] = amp * (a * s - invlam * c);
        return;
    }

    int j = idx - TABLE;                    // padded modes element
    if (j < NN * PADW) {
        int k = j / PADW;                   // source row ix in [0,96)
        int c = j % PADW;                   // padded column in [0,286)
        float mre = 0.f, mim = 0.f;
        if (c >= 95 && c < 95 + NN) {
            int iy = c - 95;
            float s, co;
            sincosf(w[k * NN + iy], &s, &co);
            float xv = x[k * NN + iy];
            mre = co * xv;                  // Re(e^{jw} x)
            mim = s  * xv;                  // Im(e^{jw} x)
        }
        int o = (k >> 1) * PROW + (c << 1) + (k & 1);   // row-pair interleave
        m_re2[o] = mre;
        m_im2[o] = mim;
    }
}

// ------------------- main: per-dy Toeplitz GEMMs via WMMA -------------------
// block = 1 wave (32 threads); grid = NTILE * SLICES.
// Each wave owns one 16x16 output tile and a slice of dy diagonals.
__global__ void __launch_bounds__(32)
dol_corr_kernel(const float* __restrict__ ws, float* __restrict__ partial)
{
    const float* g_re  = ws + WS_GRE;
    const float* g_im  = ws + WS_GIM;
    const float* m_re2 = ws + WS_MRE;
    const float* m_im2 = ws + WS_MIM;

    const int tile  = blockIdx.x % NTILE;
    const int slice = blockIdx.x / NTILE;
    const int jx0 = (tile / TDIM) * 16;
    const int jy0 = (tile % TDIM) * 16;

    const int lane   = threadIdx.x;      // 0..31
    const int lh     = lane >> 4;        // half-wave: selects K pair / M+8
    const int lane16 = lane & 15;        // A: row (M);  B/C/D: column (N)

    v8f acc_re = {0.f, 0.f, 0.f, 0.f, 0.f, 0.f, 0.f, 0.f};
    v8f acc_im = {0.f, 0.f, 0.f, 0.f, 0.f, 0.f, 0.f, 0.f};

    const int dyStart = slice * DY_PER;
    const int dyEnd   = (dyStart + DY_PER < DMAX) ? (dyStart + DY_PER) : DMAX;
    const int abase   = 95 - jx0 - lane16;               // dxi = abase + k

    for (int dyi = dyStart; dyi < dyEnd; ++dyi) {
        // padded column index: col95 = (jy + dy) + 95 = jy0 + lane16 + dyi
        const int col95 = jy0 + lane16 + dyi;            // always in [0,285]
        const float* gr = g_re + dyi * DMAX;             // contiguous in dxi
        const float* gi = g_im + dyi * DMAX;
        const float* br = m_re2 + (col95 << 1) + (lh * PROW);
        const float* bi = m_im2 + (col95 << 1) + (lh * PROW);

#pragma unroll 4
        for (int ix = 0; ix < NN; ix += 4) {
            const int k0 = ix + 2 * lh;                  // this lane's K pair
            const int ai = abase + k0;                   // in [0,190] always

            v2f a_re, a_im, a_imn, b_re, b_im;
            a_re.x = gr[ai];     a_re.y = gr[ai + 1];    // merges to b64
            a_im.x = gi[ai];     a_im.y = gi[ai + 1];
            a_imn.x = -a_im.x;   a_imn.y = -a_im.y;      // f32 WMMA has no neg_a

            // row-pair interleave: one aligned, coalesced b64 per fragment
            const int bo = (ix >> 1) * PROW;             // (ix/2 + lh)*572 via br
            b_re = *reinterpret_cast<const v2f*>(br + bo);
            b_im = *reinterpret_cast<const v2f*>(bi + bo);

            // complex MAC: out_re += Ar*Br - Ai*Bi ; out_im += Ar*Bi + Ai*Br
            acc_re = __builtin_amdgcn_wmma_f32_16x16x4_f32(
                false, a_re,  false, b_re, (short)0, acc_re, false, false);
            acc_re = __builtin_amdgcn_wmma_f32_16x16x4_f32(
                false, a_imn, false, b_im, (short)0, acc_re, false, false);
            acc_im = __builtin_amdgcn_wmma_f32_16x16x4_f32(
                false, a_re,  false, b_im, (short)0, acc_im, false, false);
            acc_im = __builtin_amdgcn_wmma_f32_16x16x4_f32(
                false, a_im,  false, b_re, (short)0, acc_im, false, false);
        }
    }

    float* p = partial + (size_t)(slice * NTILE + tile) * 2 * 256;
#pragma unroll
    for (int v = 0; v < 8; ++v) {
        const int elem = (v + 8 * lh) * 16 + lane16;     // (M)*16 + N
        p[elem]       = acc_re[v];
        p[256 + elem] = acc_im[v];
    }
}

// ---------------- deterministic split-K reduce + dA scaling -----------------
__global__ void dol_reduce_kernel(const float* __restrict__ partial,
                                  float* __restrict__ out)
{
    int j = blockIdx.x * blockDim.x + threadIdx.x;
    if (j >= M_TOT) return;
    const int jx = j / NN, jy = j % NN;
    const int tile = (jx >> 4) * TDIM + (jy >> 4);
    const int elem = (jx & 15) * 16 + (jy & 15);

    float re = 0.f, im = 0.f;
    for (int s = 0; s < SLICES; ++s) {
        const float* p = partial + (size_t)(s * NTILE + tile) * 2 * 256;
        re += p[elem];
        im += p[256 + elem];
    }
    const float dA = 1.55e-6f * 1.55e-6f;
    out[2 * j]     = re * dA;   // complex64: interleaved re, im
    out[2 * j + 1] = im * dA;
}

// ---------------------------------------------------------------------------
extern "C" void kernel_launch(void* const* d_in, const int* in_sizes, int n_in,
                              void* d_out, int out_size, void* d_ws, size_t ws_size,
                              hipStream_t stream)
{
    (void)in_sizes; (void)n_in; (void)out_size; (void)ws_size;

    const float* x = (const float*)d_in[0];       // (N,N) float32
    const float* w = (const float*)d_in[1];       // (N,N) float32
    // d_in[2], d_in[3]: coords — implied by the uniform pitch grid

    float* ws      = (float*)d_ws;
    float* partial = ws + WS_PART;

    const int setupThreads = TABLE + NN * PADW;   // 36481 + 27456 = 63937
    dol_setup_kernel<<<(setupThreads + 255) / 256, 256, 0, stream>>>(x, w, ws);
    dol_corr_kernel<<<NTILE * SLICES, 32, 0, stream>>>(ws, partial);
    dol_reduce_kernel<<<(M_TOT + 255) / 256, 256, 0, stream>>>(partial, (float*)d_out);
}